// GptOssTopKRouter_19774029431103
// MI455X (gfx1250) — compile-verified
//
#include <hip/hip_runtime.h>

#define HIDDEN 2880
#define NEXP 32
#define TOPK 4
#define TOKENS 16384

#define WAVES_PER_TILE 4
#define TILES_PER_BLOCK 2
#define WAVES_PER_BLOCK (WAVES_PER_TILE * TILES_PER_BLOCK)   // 8
#define THREADS_PER_BLOCK (WAVES_PER_BLOCK * 32)             // 256
#define TOKENS_PER_BLOCK (TILES_PER_BLOCK * 16)              // 32
#define KCHUNK (HIDDEN / WAVES_PER_TILE)                     // 720
#define ROWSTRIDE 33                                         // pad to avoid LDS bank conflicts

typedef __attribute__((ext_vector_type(2))) float v2f;
typedef __attribute__((ext_vector_type(4))) float v4f;
typedef __attribute__((ext_vector_type(8))) float v8f;
typedef __attribute__((ext_vector_type(4))) int   v4i;

__global__ __launch_bounds__(THREADS_PER_BLOCK)
void GptOssTopKRouter_kernel(const float* __restrict__ x,
                             const float* __restrict__ w,
                             const float* __restrict__ bias,
                             float* __restrict__ scores,
                             int* __restrict__ topidx) {
    __shared__ float lds[WAVES_PER_BLOCK][16 * ROWSTRIDE];

    const int tid    = threadIdx.x;
    const int wave   = tid >> 5;
    const int lane   = tid & 31;
    const int tile   = wave / WAVES_PER_TILE;   // 0..1 : which 16-token tile
    const int kslice = wave % WAVES_PER_TILE;   // 0..3 : which H chunk
    const int m      = lane & 15;               // A row (token) / B col (expert)
    const int kc     = lane >> 4;               // K half within the 4-wide step

    const int tok0 = blockIdx.x * TOKENS_PER_BLOCK + tile * 16;

    const float* __restrict__ xrow = x + (size_t)(tok0 + m) * HIDDEN;
    const float* __restrict__ w0   = w + (size_t)m * HIDDEN;          // experts 0..15
    const float* __restrict__ w1   = w + (size_t)(m + 16) * HIDDEN;   // experts 16..31

    const int h_begin = kslice * KCHUNK;
    const int h_end   = h_begin + KCHUNK;
    const int off     = 2 * kc;

    v8f acc0 = {};
    v8f acc1 = {};

    // K loop: exact fp32 WMMA, D = A(16x4) * B(4x16) + C
    #pragma unroll 4
    for (int h = h_begin; h < h_end; h += 4) {
        v2f a  = *(const v2f*)(xrow + h + off);
        v2f b0 = *(const v2f*)(w0   + h + off);
        v2f b1 = *(const v2f*)(w1   + h + off);
        acc0 = __builtin_amdgcn_wmma_f32_16x16x4_f32(
            /*neg_a=*/false, a, /*neg_b=*/false, b0,
            /*c_mod=*/(short)0, acc0, /*reuse_a=*/false, /*reuse_b=*/false);
        acc1 = __builtin_amdgcn_wmma_f32_16x16x4_f32(
            false, a, false, b1, (short)0, acc1, false, false);
    }

    // C/D layout: lane L, VGPR v holds D[M = v + 8*(L>>4)][N = L&15]
    float* __restrict__ my = lds[wave];
    #pragma unroll
    for (int v = 0; v < 8; ++v) {
        const int t = v + 8 * kc;
        my[t * ROWSTRIDE + m]      = acc0[v];
        my[t * ROWSTRIDE + m + 16] = acc1[v];
    }
    __syncthreads();

    // Phase 2: one lane per token. Reduce the 4 K-slice partials + bias,
    // top-4 (strict '>' scan => lowest-index tie-break like jax.lax.top_k),
    // softmax over the 4, emit full 32-float row + int4 indices.
    if (kslice == 0 && lane < 16) {
        const int t     = lane;
        const int wbase = tile * WAVES_PER_TILE;

        float logit[NEXP];
        #pragma unroll
        for (int e = 0; e < NEXP; ++e) {
            float s = bias[e];
            #pragma unroll
            for (int p = 0; p < WAVES_PER_TILE; ++p)
                s += lds[wbase + p][t * ROWSTRIDE + e];
            logit[e] = s;
        }

        int   idx[TOPK];
        float val[TOPK];
        unsigned taken = 0u;
        #pragma unroll
        for (int j = 0; j < TOPK; ++j) {
            float best = -3.402823466e+38f;
            int   bi   = 0;
            #pragma unroll
            for (int e = 0; e < NEXP; ++e) {
                const bool free = ((taken >> e) & 1u) == 0u;
                if (free && logit[e] > best) { best = logit[e]; bi = e; }
            }
            val[j] = best;
            idx[j] = bi;
            taken |= (1u << bi);
        }

        // softmax over the 4 selected logits (val[0] is the max)
        float ex[TOPK];
        float sum = 0.0f;
        #pragma unroll
        for (int j = 0; j < TOPK; ++j) { ex[j] = __expf(val[j] - val[0]); sum += ex[j]; }
        const float inv = 1.0f / sum;

        float* __restrict__ orow = scores + (size_t)(tok0 + t) * NEXP;
        #pragma unroll
        for (int c = 0; c < NEXP / 4; ++c) {
            v4f chunk;
            #pragma unroll
            for (int q = 0; q < 4; ++q) {
                const int e = c * 4 + q;
                float pv = 0.0f;
                #pragma unroll
                for (int j = 0; j < TOPK; ++j)
                    pv = (idx[j] == e) ? (ex[j] * inv) : pv;
                chunk[q] = pv;
            }
            *(v4f*)(orow + c * 4) = chunk;
        }

        v4i iv = { idx[0], idx[1], idx[2], idx[3] };
        *(v4i*)(topidx + (size_t)(tok0 + t) * TOPK) = iv;
    }
}

extern "C" void kernel_launch(void* const* d_in, const int* in_sizes, int n_in,
                              void* d_out, int out_size, void* d_ws, size_t ws_size,
                              hipStream_t stream) {
    (void)in_sizes; (void)n_in; (void)d_ws; (void)ws_size; (void)out_size;

    const float* x    = (const float*)d_in[0];   // [T, H]
    const float* w    = (const float*)d_in[1];   // [E, H]
    const float* bias = (const float*)d_in[2];   // [E]

    float* scores = (float*)d_out;                                   // [T, E] fp32
    int*   topidx = (int*)((float*)d_out + (size_t)TOKENS * NEXP);   // [T, K] int32

    dim3 grid(TOKENS / TOKENS_PER_BLOCK);   // 512 blocks
    dim3 block(THREADS_PER_BLOCK);          // 256 threads = 8 waves
    hipLaunchKernelGGL(GptOssTopKRouter_kernel, grid, block, 0, stream,
                       x, w, bias, scores, topidx);
}